// SerializedPooling_13529146982504
// MI455X (gfx1250) — compile-verified
//
#include <hip/hip_runtime.h>
#include <math.h>

#define NPTS    524288
#define INC     128
#define OUTC    256
#define KSEG    65536
#define PDSHIFT 3
#define BN_EPS  0.001f

typedef __attribute__((ext_vector_type(2))) float v2f;
typedef __attribute__((ext_vector_type(8))) float v8f;
typedef __attribute__((ext_vector_type(4))) int   i32x4;
typedef __attribute__((address_space(1))) i32x4   gi32x4;   // global (AS1)
typedef __attribute__((address_space(3))) i32x4   li32x4;   // LDS (AS3)

// ---------------------------------------------------------------------------
// K1: scatter points into their bucket's 8 slots (arrival order, fixed later)
// ---------------------------------------------------------------------------
__global__ void sp_scatter(const long long* __restrict__ sc0,
                           unsigned* __restrict__ cnt,
                           unsigned* __restrict__ members) {
    int i = blockIdx.x * blockDim.x + threadIdx.x;            // grid = N/256
    unsigned c = (unsigned)(sc0[i] >> PDSHIFT);               // bucket in [0,K)
    unsigned slot = atomicAdd(&cnt[c], 1u);                   // exactly 8 per bucket
    members[c * 8u + slot] = (unsigned)i;
}

// ---------------------------------------------------------------------------
// K2: per bucket: stable-sort 8 members, head index, coord mean, pooled_code,
//     pooled_grid, identity order/inverse for code row 0
// ---------------------------------------------------------------------------
__global__ void sp_bucket_finalize(const long long* __restrict__ sc,     // (3,N)
                                   const float* __restrict__ coord,      // (N,3)
                                   const int* __restrict__ grid_coord,   // (N,3)
                                   unsigned* __restrict__ indices,       // in-place sort of members
                                   float* __restrict__ coord_mean,       // (K,3)
                                   long long* __restrict__ pooled_code,  // (3,K)
                                   long long* __restrict__ pooled_order, // (3,K)
                                   long long* __restrict__ pooled_inverse,
                                   int* __restrict__ pooled_grid) {      // (K,3)
    int c = blockIdx.x * blockDim.x + threadIdx.x;            // grid = K/256
    unsigned m[8];
#pragma unroll
    for (int j = 0; j < 8; ++j) m[j] = indices[c * 8 + j];
    // bubble network (fully unrolled) -> ascending = stable order by index
#pragma unroll
    for (int a = 0; a < 8; ++a)
#pragma unroll
        for (int b2 = 0; b2 < 7 - a; ++b2)
            if (m[b2] > m[b2 + 1]) { unsigned t = m[b2]; m[b2] = m[b2 + 1]; m[b2 + 1] = t; }
    float sx = 0.f, sy = 0.f, sz = 0.f;
#pragma unroll
    for (int j = 0; j < 8; ++j) {
        indices[c * 8 + j] = m[j];
        sx += coord[(size_t)m[j] * 3 + 0];
        sy += coord[(size_t)m[j] * 3 + 1];
        sz += coord[(size_t)m[j] * 3 + 2];
    }
    coord_mean[c * 3 + 0] = sx * 0.125f;
    coord_mean[c * 3 + 1] = sy * 0.125f;
    coord_mean[c * 3 + 2] = sz * 0.125f;
    unsigned head = m[0];
#pragma unroll
    for (int r = 0; r < 3; ++r)
        pooled_code[(size_t)r * KSEG + c] = sc[(size_t)r * NPTS + head] >> PDSHIFT;
    // code row 0 at heads is exactly c -> argsort is identity
    pooled_order[c]   = (long long)c;
    pooled_inverse[c] = (long long)c;
    pooled_grid[c * 3 + 0] = grid_coord[(size_t)head * 3 + 0] >> 1;
    pooled_grid[c * 3 + 1] = grid_coord[(size_t)head * 3 + 1] >> 1;
    pooled_grid[c * 3 + 2] = grid_coord[(size_t)head * 3 + 2] >> 1;
}

// ---------------------------------------------------------------------------
// K3: counting-sort scatter of head codes (values in [0,K), <=8 per bin)
// ---------------------------------------------------------------------------
__global__ void sp_head_scatter(const long long* __restrict__ pooled_code_r,
                                unsigned* __restrict__ cnt,
                                unsigned* __restrict__ members) {
    int c = blockIdx.x * blockDim.x + threadIdx.x;            // grid = K/256
    unsigned v = (unsigned)pooled_code_r[c];
    unsigned slot = atomicAdd(&cnt[v], 1u);
    members[v * 8u + slot] = (unsigned)c;
}

// ---------------------------------------------------------------------------
// K4: exclusive scan of 65536 bin counts, one 1024-thread block
// ---------------------------------------------------------------------------
__global__ void sp_scan(const unsigned* __restrict__ cnt, unsigned* __restrict__ offs) {
    __shared__ unsigned sbuf[1024];
    const int t = threadIdx.x;
    const int CH = KSEG / 1024;                               // 64 bins / thread
    unsigned base = (unsigned)t * CH;
    unsigned s = 0;
    for (int j = 0; j < CH; ++j) s += cnt[base + j];
    sbuf[t] = s;
    __syncthreads();
    unsigned mine = s;
    for (int d = 1; d < 1024; d <<= 1) {
        unsigned v = (t >= d) ? sbuf[t - d] : 0u;
        __syncthreads();
        sbuf[t] += v;
        __syncthreads();
    }
    unsigned run = sbuf[t] - mine;                            // exclusive prefix
    for (int j = 0; j < CH; ++j) { offs[base + j] = run; run += cnt[base + j]; }
}

// ---------------------------------------------------------------------------
// K5: per bin: sort members ascending (stability), emit order + inverse
// ---------------------------------------------------------------------------
__global__ void sp_emit_order(const unsigned* __restrict__ cnt,
                              const unsigned* __restrict__ members,
                              const unsigned* __restrict__ offs,
                              long long* __restrict__ order_r,
                              long long* __restrict__ inverse_r) {
    int v = blockIdx.x * blockDim.x + threadIdx.x;            // grid = K/256
    unsigned n = cnt[v];
    if (n == 0) return;
    unsigned m[8];
#pragma unroll
    for (int j = 0; j < 8; ++j) m[j] = (j < (int)n) ? members[v * 8u + j] : 0xFFFFFFFFu;
#pragma unroll
    for (int a = 0; a < 8; ++a)
#pragma unroll
        for (int b2 = 0; b2 < 7 - a; ++b2)
            if (m[b2] > m[b2 + 1]) { unsigned t = m[b2]; m[b2] = m[b2 + 1]; m[b2 + 1] = t; }
    unsigned o = offs[v];
    for (unsigned j = 0; j < n; ++j) {
        unsigned c = m[j];
        order_r[o + j] = (long long)c;
        inverse_r[c]   = (long long)(o + j);
    }
}

// ---------------------------------------------------------------------------
// K6: fused gather + fp32 WMMA GEMM + per-bucket max  (the HBM-bound hot loop)
//   - W (256x128 f32 = 128 KB) staged once per workgroup into static LDS,
//     via GLOBAL_LOAD_ASYNC_TO_LDS_B128 (ASYNCcnt) when available
//   - each wave: one M-tile = 16 gathered rows = 2 buckets
//   - A fragments preloaded (32 x v2f = 64 VGPRs), reused across 16 N-tiles
//   - V_WMMA_F32_16X16X4_F32, 32 k-steps per N-tile
// ---------------------------------------------------------------------------
__global__ void __launch_bounds__(256)
sp_gemm_max(const float* __restrict__ feat,
            const float* __restrict__ W,
            const float* __restrict__ bias,
            const unsigned* __restrict__ indices,
            float* __restrict__ feat_max) {
    __shared__ float ldsW[OUTC * INC];                        // 128 KB static LDS

#if __has_builtin(__builtin_amdgcn_global_load_async_to_lds_b128)
    {
        gi32x4* gsrc = (gi32x4*)W;                            // AS1, const stripped
        li32x4* ldst = (li32x4*)ldsW;                         // AS3
        for (int i = threadIdx.x; i < OUTC * INC / 4; i += 256)
            __builtin_amdgcn_global_load_async_to_lds_b128(gsrc + i, ldst + i, 0, 0);
#if __has_builtin(__builtin_amdgcn_s_wait_asynccnt)
        __builtin_amdgcn_s_wait_asynccnt(0);
#else
        asm volatile("s_wait_asynccnt 0" ::: "memory");
#endif
    }
#else
    for (int i = threadIdx.x; i < OUTC * INC / 4; i += 256)
        ((float4*)ldsW)[i] = ((const float4*)W)[i];
#endif
    __syncthreads();

    const int wave  = threadIdx.x >> 5;
    const int lane  = threadIdx.x & 31;
    const int l16   = lane & 15;
    const int hs    = lane >> 4;                              // half-wave select
    const int mtile = blockIdx.x * 8 + wave;                  // grid = K/16 -> 32768 tiles

    // A: 16x4 f32 tile per k-step; lanes 0-15 hold K=4k+{0,1}, lanes 16-31 K=4k+{2,3}
    const unsigned row  = indices[mtile * 16 + l16];
    const float*   arow = feat + (size_t)row * INC + 2 * hs;
    v2f a[32];
#pragma unroll
    for (int ks = 0; ks < 32; ++ks) {
        float2 t = *(const float2*)(arow + 4 * ks);
        a[ks][0] = t.x; a[ks][1] = t.y;
    }

    // hoist per-column bias loads out of the WMMA loop
    float bvals[16];
#pragma unroll
    for (int nt = 0; nt < 16; ++nt) bvals[nt] = bias[nt * 16 + l16];

    const int outrow = mtile * 2 + hs;                        // bucket id for this half-wave
    for (int nt = 0; nt < 16; ++nt) {
        // B[k][n] = W[n][k]; lane holds col n0+l16, K=4k+2*hs+{0,1}
        const float* bcol = ldsW + (size_t)(nt * 16 + l16) * INC + 2 * hs;
        v8f c = {0.f, 0.f, 0.f, 0.f, 0.f, 0.f, 0.f, 0.f};
#pragma unroll
        for (int ks = 0; ks < 32; ++ks) {
            float2 t = *(const float2*)(bcol + 4 * ks);
            v2f b; b[0] = t.x; b[1] = t.y;
            c = __builtin_amdgcn_wmma_f32_16x16x4_f32(
                    /*neg_a=*/false, a[ks], /*neg_b=*/false, b,
                    /*c_mod=*/(short)0, c, /*reuse_a=*/false, /*reuse_b=*/false);
        }
        // D rows 0-7 live in VGPRs 0-7 of lanes 0-15 (bucket 2t), rows 8-15 in lanes 16-31
        float m = fmaxf(fmaxf(fmaxf(c[0], c[1]), fmaxf(c[2], c[3])),
                        fmaxf(fmaxf(c[4], c[5]), fmaxf(c[6], c[7])));
        // max(x + b) == max(x) + b for constant per-column bias
        feat_max[(size_t)outrow * OUTC + nt * 16 + l16] = m + bvals[nt];
    }
}

// ---------------------------------------------------------------------------
// K7/K8/K9: BN stats over rows, finalize, BN+exact GELU in place
// ---------------------------------------------------------------------------
__global__ void sp_colstats(const float* __restrict__ fm,
                            float* __restrict__ sums, float* __restrict__ sumsq) {
    int j  = threadIdx.x;                                     // column
    int r0 = blockIdx.x * 256;                                // grid = K/256 row-chunks
    float s = 0.f, s2 = 0.f;
    for (int r = 0; r < 256; ++r) {
        float x = fm[(size_t)(r0 + r) * OUTC + j];
        s += x; s2 += x * x;
    }
    atomicAdd(&sums[j], s);
    atomicAdd(&sumsq[j], s2);
}

__global__ void sp_bnparams(const float* __restrict__ sums, const float* __restrict__ sumsq,
                            const float* __restrict__ gamma, const float* __restrict__ beta,
                            float* __restrict__ scale, float* __restrict__ shift) {
    int j = threadIdx.x;
    float mean = sums[j] * (1.f / KSEG);
    float var  = sumsq[j] * (1.f / KSEG) - mean * mean;
    float sc   = rsqrtf(var + BN_EPS) * gamma[j];
    scale[j] = sc;
    shift[j] = beta[j] - mean * sc;
}

__global__ void sp_bn_gelu(float* __restrict__ fm,
                           const float* __restrict__ scale, const float* __restrict__ shift) {
    size_t i = (size_t)blockIdx.x * blockDim.x + threadIdx.x;
    int j = (int)(i & (OUTC - 1));
    float y = fm[i] * scale[j] + shift[j];
    fm[i] = 0.5f * y * (1.f + erff(y * 0.70710678118654752f));
}

// ---------------------------------------------------------------------------
extern "C" void kernel_launch(void* const* d_in, const int* in_sizes, int n_in,
                              void* d_out, int out_size, void* d_ws, size_t ws_size,
                              hipStream_t stream) {
    (void)in_sizes; (void)n_in; (void)out_size; (void)ws_size;
    const float*     feat       = (const float*)d_in[0];
    const float*     coord      = (const float*)d_in[1];
    const float*     W          = (const float*)d_in[2];
    const float*     bias       = (const float*)d_in[3];
    const float*     gamma      = (const float*)d_in[4];
    const float*     beta       = (const float*)d_in[5];
    const long long* sc         = (const long long*)d_in[6];
    const int*       grid_coord = (const int*)d_in[7];

    // outputs concatenated in return order (byte layout)
    char* ob = (char*)d_out;
    float*     out_feat       = (float*)ob;                                       // (K,256) f32
    float*     coord_mean     = (float*)(ob + (size_t)KSEG * OUTC * 4);           // (K,3)  f32
    long long* pooled_code    = (long long*)(ob + (size_t)KSEG * OUTC * 4
                                                + (size_t)KSEG * 3 * 4);          // (3,K) i64
    long long* pooled_order   = pooled_code  + 3 * (size_t)KSEG;                  // (3,K) i64
    long long* pooled_inverse = pooled_order + 3 * (size_t)KSEG;                  // (3,K) i64
    int*       pooled_grid    = (int*)(pooled_inverse + 3 * (size_t)KSEG);        // (K,3) i32

    // workspace layout
    char* wb = (char*)d_ws;
    size_t o = 0;
    unsigned* cntA     = (unsigned*)(wb + o); o += (size_t)KSEG * 4;
    unsigned* cntB     = (unsigned*)(wb + o); o += (size_t)KSEG * 4;
    unsigned* cntC     = (unsigned*)(wb + o); o += (size_t)KSEG * 4;
    unsigned* indices  = (unsigned*)(wb + o); o += (size_t)NPTS * 4;
    unsigned* members2 = (unsigned*)(wb + o); o += (size_t)KSEG * 8 * 4;
    unsigned* offs     = (unsigned*)(wb + o); o += (size_t)KSEG * 4;
    float*    sums     = (float*)(wb + o);    o += OUTC * 4;
    float*    sumsq    = (float*)(wb + o);    o += OUTC * 4;
    float*    scale    = (float*)(wb + o);    o += OUTC * 4;
    float*    shiftp   = (float*)(wb + o);    o += OUTC * 4;

    (void)hipMemsetAsync(cntA, 0, (size_t)KSEG * 4, stream);
    (void)hipMemsetAsync(cntB, 0, (size_t)KSEG * 4, stream);
    (void)hipMemsetAsync(cntC, 0, (size_t)KSEG * 4, stream);
    (void)hipMemsetAsync(sums, 0, (size_t)OUTC * 4 * 2, stream);  // sums + sumsq contiguous

    sp_scatter<<<NPTS / 256, 256, 0, stream>>>(sc, cntA, indices);
    sp_bucket_finalize<<<KSEG / 256, 256, 0, stream>>>(sc, coord, grid_coord, indices,
        coord_mean, pooled_code, pooled_order, pooled_inverse, pooled_grid);

    // code row 1
    sp_head_scatter<<<KSEG / 256, 256, 0, stream>>>(pooled_code + KSEG, cntB, members2);
    sp_scan<<<1, 1024, 0, stream>>>(cntB, offs);
    sp_emit_order<<<KSEG / 256, 256, 0, stream>>>(cntB, members2, offs,
        pooled_order + KSEG, pooled_inverse + KSEG);
    // code row 2 (reuses members2/offs; stream-ordered after row-1 consumers)
    sp_head_scatter<<<KSEG / 256, 256, 0, stream>>>(pooled_code + 2 * (size_t)KSEG, cntC, members2);
    sp_scan<<<1, 1024, 0, stream>>>(cntC, offs);
    sp_emit_order<<<KSEG / 256, 256, 0, stream>>>(cntC, members2, offs,
        pooled_order + 2 * (size_t)KSEG, pooled_inverse + 2 * (size_t)KSEG);

    // fused gather + WMMA GEMM + segment-max (32768 M-tiles, 8 waves/block)
    sp_gemm_max<<<KSEG / 16, 256, 0, stream>>>(feat, W, bias, indices, out_feat);

    sp_colstats<<<KSEG / 256, 256, 0, stream>>>(out_feat, sums, sumsq);
    sp_bnparams<<<1, OUTC, 0, stream>>>(sums, sumsq, gamma, beta, scale, shiftp);
    sp_bn_gelu<<<(size_t)KSEG * OUTC / 256, 256, 0, stream>>>(out_feat, scale, shiftp);
}